// RLMahjong_34342558499407
// MI455X (gfx1250) — compile-verified
//
#include <hip/hip_runtime.h>

#define ALPHA  0.99f
#define GAMMA  0.999f
#define LAMBD  0.7f

typedef __attribute__((ext_vector_type(2))) float v2f;
typedef __attribute__((ext_vector_type(8))) float v8f;

// ---------------------------------------------------------------------------
// Kernel 1: fused per-row pass.
//   - one wave32 per 16-row tile
//   - v_t = sum(pi_t * q_t), v_tp1 = sum(pi_tp1 * q_tar_tp1) via chained
//     V_WMMA_F32_16X16X4_F32 with an all-ones B matrix (layout-invariant),
//     A matrix = elementwise product laid out per ISA 16x4 f32 A layout.
//   - writes advantages = (1-ALPHA)*(q_t - v_t)
//   - writes per-t scan coefficients to workspace:
//       wsA = done ? 0 : GAMMA*c        (carry multiplier)
//       wsB = LAMBD*GAMMA*rho*td        (carry additive)
//       wsE = est, wsM = done?0:1, wsD = ALPHA*(q_t_a - v_t)
// ---------------------------------------------------------------------------
__global__ __launch_bounds__(256)
void rlm_prep(const float* __restrict__ q, const float* __restrict__ q_tar,
              const float* __restrict__ pi, const int* __restrict__ a_t,
              const float* __restrict__ r_t, const float* __restrict__ mu,
              const int* __restrict__ done_t,
              float* __restrict__ adv,
              float* __restrict__ wsA, float* __restrict__ wsB,
              float* __restrict__ wsE, float* __restrict__ wsM,
              float* __restrict__ wsD)
{
    __shared__ float lds_v[8 * 16];
    __shared__ float lds_vp[8 * 16];

    const int lane = threadIdx.x & 31;
    const int wave = threadIdx.x >> 5;
    const int tile = blockIdx.x * 8 + wave;     // 16 rows (timesteps) per tile
    const int t0   = tile * 16;
    const int row  = lane & 15;                 // M index (ISA A-matrix layout)
    const int colbase = (lane >> 4) << 1;       // low half: K%4 in {0,1}; high: {2,3}

    const float* qrow   = q     + (size_t)(t0 + row) * 64 + colbase;
    const float* pirow  = pi    + (size_t)(t0 + row) * 64 + colbase;
    const float* qtrow  = q_tar + (size_t)(t0 + row + 1) * 64 + colbase;
    const float* piprow = pi    + (size_t)(t0 + row + 1) * 64 + colbase;

    const v2f ones = {1.0f, 1.0f};
    v8f acc_v  = {};
    v8f acc_vp = {};
    v2f qv[16];                                  // keep q for the advantage pass

    #pragma unroll
    for (int s = 0; s < 16; ++s) {               // K = 4*s .. 4*s+3
        v2f qd = *(const v2f*)(qrow  + 4 * s);
        v2f pd = *(const v2f*)(pirow + 4 * s);
        qv[s] = qd;
        v2f x = qd * pd;
        acc_v = __builtin_amdgcn_wmma_f32_16x16x4_f32(
            false, x, false, ones, (short)0, acc_v, false, false);
        v2f qtd = *(const v2f*)(qtrow  + 4 * s);
        v2f ppd = *(const v2f*)(piprow + 4 * s);
        v2f x2 = qtd * ppd;
        acc_vp = __builtin_amdgcn_wmma_f32_16x16x4_f32(
            false, x2, false, ones, (short)0, acc_vp, false, false);
    }

    // D layout: VGPR r holds v[r] in lanes 0-15 and v[8+r] in lanes 16-31.
    // Select my element (compile-time indices only), deposit to LDS.
    float selv = acc_v[0], selvp = acc_vp[0];
    #pragma unroll
    for (int j = 1; j < 8; ++j) {
        bool pick = ((row & 7) == j);
        selv  = pick ? acc_v[j]  : selv;
        selvp = pick ? acc_vp[j] : selvp;
    }
    float* myv  = lds_v  + wave * 16;
    float* myvp = lds_vp + wave * 16;
    if ((lane & 15) < 8) {
        // lane j (<8) holds v[j]; lane 16+j holds v[8+j]
        int idx = (lane & 7) + ((lane >> 4) << 3);
        myv[idx]  = selv;
        myvp[idx] = selvp;
    }
    __syncthreads();   // uniform; orders LDS store -> load

    const float vt = myv[row];

    // advantages = (1-ALPHA) * (q_t - v_t), same interleaved layout as loads
    float* advrow = adv + (size_t)(t0 + row) * 64 + colbase;
    const float ascale = 1.0f - ALPHA;
    #pragma unroll
    for (int s = 0; s < 16; ++s) {
        v2f o = (qv[s] - vt) * ascale;
        *(v2f*)(advrow + 4 * s) = o;
    }

    // Per-timestep scalar work: lanes 0..15 handle rows 0..15.
    if (lane < 16) {
        const int t = t0 + lane;
        const int a = a_t[t];
        const float q_t_a = q [(size_t)t * 64 + a];
        const float pi_a  = pi[(size_t)t * 64 + a];
        const float mu_a  = mu[(size_t)t * 64 + a];
        const float r     = r_t[t];
        const int   d     = done_t[t];
        const float df    = (float)d;
        const float vp1   = myvp[lane];
        const float vv    = myv[lane];

        const float est = r + (1.0f - df) * GAMMA * vp1;
        const float td  = est - q_t_a;
        const float rho = pi_a / mu_a;
        const float cc  = LAMBD * fminf(fmaxf(rho, 0.0f), 1.0f);

        wsB[t] = LAMBD * GAMMA * rho * td;
        wsA[t] = d ? 0.0f : GAMMA * cc;
        wsM[t] = d ? 0.0f : 1.0f;
        wsE[t] = est;
        wsD[t] = ALPHA * (q_t_a - vv);
    }
}

// ---------------------------------------------------------------------------
// Kernel 2: per-chunk reverse affine composition.  carry_t = B_t + A_t*carry_{t+1}
// Composition over chunk [lo,hi): F = f_lo ∘ ... ∘ f_{hi-1} as (Ac, Bc).
// ---------------------------------------------------------------------------
__global__ __launch_bounds__(256)
void rlm_compose(const float* __restrict__ wsA, const float* __restrict__ wsB,
                 float* __restrict__ chA, float* __restrict__ chB, int S)
{
    const int c = blockIdx.x * blockDim.x + threadIdx.x;
    const int base = c * S;
    float Ac = 1.0f, Bc = 0.0f;
    for (int i = S - 1; i >= 0; --i) {
        const float a = wsA[base + i];
        const float b = wsB[base + i];
        Bc = b + a * Bc;     // f_t ∘ (accumulated right part)
        Ac = a * Ac;
    }
    chA[c] = Ac;
    chB[c] = Bc;
}

// ---------------------------------------------------------------------------
// Kernel 3: single-block reverse inclusive scan of chunk affines (Hillis-Steele
// in LDS), then carryIn[c] = (inclusive over c+1..C-1) applied to y=0 = B.
// ---------------------------------------------------------------------------
__global__ __launch_bounds__(1024)
void rlm_chunkscan(const float* __restrict__ chA, const float* __restrict__ chB,
                   float* __restrict__ carryIn, int C)
{
    __shared__ float sA[1024];
    __shared__ float sB[1024];
    const int i = threadIdx.x;
    if (i < C) { sA[i] = chA[i]; sB[i] = chB[i]; }
    __syncthreads();
    for (int off = 1; off < C; off <<= 1) {
        float a = 0.f, b = 0.f, a2 = 0.f, b2 = 0.f;
        const bool ok = (i + off) < C && i < C;
        if (ok) { a = sA[i]; b = sB[i]; a2 = sA[i + off]; b2 = sB[i + off]; }
        __syncthreads();
        if (ok) { sA[i] = a * a2; sB[i] = b + a * b2; }   // left ∘ right
        __syncthreads();
    }
    if (i < C) carryIn[i] = (i + 1 < C) ? sB[i + 1] : 0.0f;
}

// ---------------------------------------------------------------------------
// Kernel 4: finalize.  Walk each chunk backwards from its carry-in; emit
// targets_q[t] = est + m*carry_{t+1} + delta, then advance the carry.
// ---------------------------------------------------------------------------
__global__ __launch_bounds__(256)
void rlm_finalize(const float* __restrict__ wsA, const float* __restrict__ wsB,
                  const float* __restrict__ wsE, const float* __restrict__ wsM,
                  const float* __restrict__ wsD, const float* __restrict__ carryIn,
                  float* __restrict__ targets, int S)
{
    const int c = blockIdx.x * blockDim.x + threadIdx.x;
    const int base = c * S;
    float y = carryIn[c];                       // carry entering from the right
    for (int i = S - 1; i >= 0; --i) {
        const int t = base + i;
        const float g = wsE[t] + wsM[t] * y;    // g = est + (done?0:carry_{t+1})
        targets[t] = g + wsD[t];
        y = wsB[t] + wsA[t] * y;                // carry_t
    }
}

// ---------------------------------------------------------------------------
extern "C" void kernel_launch(void* const* d_in, const int* in_sizes, int n_in,
                              void* d_out, int out_size, void* d_ws, size_t ws_size,
                              hipStream_t stream)
{
    const float* q      = (const float*)d_in[0];
    const float* q_tar  = (const float*)d_in[1];
    const float* pi     = (const float*)d_in[2];
    const int*   a_t    = (const int*)  d_in[3];
    const float* r_t    = (const float*)d_in[4];
    const float* mu     = (const float*)d_in[5];
    const int*   done_t = (const int*)  d_in[6];

    const int T = in_sizes[4];                 // r_t has T elements (262144)
    float* targets = (float*)d_out;            // [T]
    float* adv     = targets + T;              // [T*64]

    float* ws  = (float*)d_ws;
    float* wsA = ws;
    float* wsB = ws + (size_t)T;
    float* wsE = ws + (size_t)2 * T;
    float* wsM = ws + (size_t)3 * T;
    float* wsD = ws + (size_t)4 * T;
    const int S = 256;                         // chunk length
    const int C = T / S;                       // 1024 chunks
    float* chA     = ws + (size_t)5 * T;
    float* chB     = chA + C;
    float* carryIn = chB + C;

    const int tiles   = T / 16;                // 16384 wave-tiles
    const int blocks1 = tiles / 8;             // 8 waves per 256-thread block

    rlm_prep<<<blocks1, 256, 0, stream>>>(q, q_tar, pi, a_t, r_t, mu, done_t,
                                          adv, wsA, wsB, wsE, wsM, wsD);
    rlm_compose<<<C / 256, 256, 0, stream>>>(wsA, wsB, chA, chB, S);
    rlm_chunkscan<<<1, 1024, 0, stream>>>(chA, chB, carryIn, C);
    rlm_finalize<<<C / 256, 256, 0, stream>>>(wsA, wsB, wsE, wsM, wsD, carryIn,
                                              targets, S);
}